// NCC_730144441164
// MI455X (gfx1250) — compile-verified
//
#include <hip/hip_runtime.h>

// ---------------------------------------------------------------------------
// Local NCC loss, fused one-pass tile kernel for MI455X (gfx1250).
//  vol shape (B=2, 1, D=160, H=192, W=160), window 9^3, 'SAME' zero padding.
//  Each workgroup: 16^3 output tile from a 24^3 halo tile staged in LDS via
//  async global->LDS loads; separable 9-tap sums (W, H in VALU; D via
//  v_wmma_f32_16x16x4_f32 as Out[16x256] = Band[16x24] x HP[24x256]);
//  per-block partial sum -> d_ws; tiny reduce kernel -> d_out[0] = -mean(cc).
// ---------------------------------------------------------------------------

#define BB 2
#define DD 160
#define HH 192
#define WW 160

#define TQ 16            // output tile edge
#define TI 24            // input tile edge (halo 4 each side)
#define NTHREADS 256

// LDS layout (floats):
//  region1 [0 .. 30719]      : stage0/A: sI(13824) + sJ(13824); stage B out: hpass 5*24*16*16 = 30720
//  region2 [30720 .. 76799]  : stage A out: wpass 5*24*24*16 = 46080; stage C out: Sbuf 5*16*256 = 20480
#define REGION1_F 30720
#define REGION2_F 46080
#define SMEM_FLOATS (REGION1_F + REGION2_F)   // 76800 floats = 307200 B (<= 320KB/WGP)

typedef __attribute__((ext_vector_type(2))) float v2f;
typedef __attribute__((ext_vector_type(8))) float v8f;
typedef __attribute__((ext_vector_type(4))) int   v4i;

typedef __attribute__((address_space(1))) v4i* gas1_v4i;
typedef __attribute__((address_space(3))) v4i* las3_v4i;
typedef __attribute__((address_space(1))) int* gas1_i32;
typedef __attribute__((address_space(3))) int* las3_i32;

#if defined(__has_builtin)
#  if __has_builtin(__builtin_amdgcn_global_load_async_to_lds_b128)
#    define HAVE_ASYNC_B128 1
#  endif
#  if __has_builtin(__builtin_amdgcn_global_load_async_to_lds_b32)
#    define HAVE_ASYNC_B32 1
#  endif
#  if __has_builtin(__builtin_amdgcn_wmma_f32_16x16x4_f32)
#    define HAVE_WMMA_F32_4 1
#  endif
#  if __has_builtin(__builtin_amdgcn_s_wait_asynccnt)
#    define WAIT_ASYNC0() __builtin_amdgcn_s_wait_asynccnt(0)
#  endif
#endif
#ifndef WAIT_ASYNC0
#  define WAIT_ASYNC0() asm volatile("s_wait_asynccnt 0" ::: "memory")
#endif

__device__ __forceinline__ void copy16_g2l(const float* gsrc, float* ldst) {
  void* g = (void*)gsrc;   // drop const; generic pointers
  void* l = (void*)ldst;
#if defined(HAVE_ASYNC_B128)
  __builtin_amdgcn_global_load_async_to_lds_b128(
      (gas1_v4i)(v4i*)g, (las3_v4i)(v4i*)l, 0, 0);
#elif defined(HAVE_ASYNC_B32)
  __builtin_amdgcn_global_load_async_to_lds_b32((gas1_i32)((int*)g + 0), (las3_i32)((int*)l + 0), 0, 0);
  __builtin_amdgcn_global_load_async_to_lds_b32((gas1_i32)((int*)g + 1), (las3_i32)((int*)l + 1), 0, 0);
  __builtin_amdgcn_global_load_async_to_lds_b32((gas1_i32)((int*)g + 2), (las3_i32)((int*)l + 2), 0, 0);
  __builtin_amdgcn_global_load_async_to_lds_b32((gas1_i32)((int*)g + 3), (las3_i32)((int*)l + 3), 0, 0);
#else
  *(float4*)ldst = *(const float4*)gsrc;   // sync fallback
#endif
}

__global__ __launch_bounds__(NTHREADS) void ncc_tile_kernel(
    const float* __restrict__ Ig, const float* __restrict__ Jg,
    float* __restrict__ partial) {
  extern __shared__ float smem[];
  float* sI    = smem;                 // 13824 f
  float* sJ    = smem + 13824;         // 13824 f
  float* hpass = smem;                 // 30720 f (aliases sI/sJ, used after stage A)
  float* wpass = smem + REGION1_F;     // 46080 f
  float* Sbuf  = smem + REGION1_F;     // 20480 f (aliases wpass, used after stage B)

  const int tid = threadIdx.x;
  const int b  = blockIdx.z / (DD / TQ);
  const int dt = blockIdx.z % (DD / TQ);
  const int d0 = dt * TQ - 4;
  const int h0 = blockIdx.y * TQ - 4;
  const int w0 = blockIdx.x * TQ - 4;

  // ---- Stage 0a: zero halo tiles (OOB must read as 0 for SAME padding) ----
  {
    float4 z; z.x = 0.f; z.y = 0.f; z.z = 0.f; z.w = 0.f;
    float4* p = (float4*)smem;                 // region1 = sI+sJ = 27648 f = 6912 float4
    for (int i = tid; i < 6912; i += NTHREADS) p[i] = z;
  }
  __syncthreads();

  // ---- Stage 0b: async global -> LDS halo tile loads (16B granules) ----
  // One 24-float row = 6 granules; OOB never straddles a granule (W mult of 4,
  // w0 = 16*bx - 4 keeps every granule 16B-aligned in global and LDS).
  for (int c = tid; c < 576 * 6; c += NTHREADS) {
    const int row = c / 6, ch = c % 6;
    const int ld = row / TI, lh = row % TI;
    const int gd = d0 + ld, gh = h0 + lh, gw = w0 + ch * 4;
    if ((unsigned)gd < (unsigned)DD && (unsigned)gh < (unsigned)HH &&
        (unsigned)gw < (unsigned)WW) {
      const long gidx = (((long)b * DD + gd) * HH + gh) * (long)WW + gw;
      const int  lidx = row * TI + ch * 4;
      copy16_g2l(Ig + gidx, sI + lidx);
      copy16_g2l(Jg + gidx, sJ + lidx);
    }
  }
  WAIT_ASYNC0();
  __syncthreads();

  // ---- Stage A: 9-tap sums along W for 5 fields; (24,24,24)->(24,24,16) ----
  for (int t = tid; t < TI * TI * TQ; t += NTHREADS) {   // 9216
    const int lwo = t & 15;
    const int rh  = t >> 4;
    const int lh  = rh % TI, ld = rh / TI;
    const float* pI = sI + (ld * TI + lh) * TI + lwo;
    const float* pJ = sJ + (ld * TI + lh) * TI + lwo;
    float a0 = 0.f, a1 = 0.f, a2 = 0.f, a3 = 0.f, a4 = 0.f;
#pragma unroll
    for (int k = 0; k < 9; ++k) {
      const float x = pI[k], y = pJ[k];
      a0 += x; a1 += y; a2 += x * x; a3 += y * y; a4 += x * y;
    }
    wpass[t]            = a0;
    wpass[9216 + t]     = a1;
    wpass[2 * 9216 + t] = a2;
    wpass[3 * 9216 + t] = a3;
    wpass[4 * 9216 + t] = a4;
  }
  __syncthreads();

  // ---- Stage B: 9-tap sums along H; (24,24,16)->(24,16,16) x5 ----
  for (int t = tid; t < 5 * TI * TQ * TQ; t += NTHREADS) {  // 30720
    const int q = t / 6144, r = t % 6144;
    const int lw = r & 15, lho = (r >> 4) & 15, ld = r >> 8;
    const float* p = wpass + q * 9216 + (ld * TI + lho) * TQ + lw;
    float s = 0.f;
#pragma unroll
    for (int k = 0; k < 9; ++k) s += p[k * TQ];
    hpass[t] = s;   // layout: q*6144 + ld*256 + lho*16 + lw
  }
  __syncthreads();

  // ---- Stage C: 9-tap sums along D via WMMA: Out[16x256] = Band[16x24] x HP[24x256]
  const int lane = tid & 31, wid = tid >> 5;
  const int mrow = lane & 15, half = lane >> 4;
#if defined(HAVE_WMMA_F32_4)
  for (int work = wid; work < 5 * 16; work += 8) {   // uniform per wave: EXEC all-1s
    const int q = work / 16, nc = work % 16;
    const int col = nc * 16 + mrow;                  // N index for B/C/D
    const float* bq = hpass + q * 6144;
    v8f acc = {0.f, 0.f, 0.f, 0.f, 0.f, 0.f, 0.f, 0.f};
#pragma unroll
    for (int kc = 0; kc < 6; ++kc) {
      const int ka = 4 * kc + 2 * half;              // A/B lane K base (v + 2*half)
      v2f A, Bv;
      A.x = ((unsigned)(ka - mrow) <= 8u) ? 1.0f : 0.0f;         // band: m <= k <= m+8
      A.y = ((unsigned)(ka + 1 - mrow) <= 8u) ? 1.0f : 0.0f;
      Bv.x = bq[ka * 256 + col];
      Bv.y = bq[(ka + 1) * 256 + col];
      acc = __builtin_amdgcn_wmma_f32_16x16x4_f32(
          false, A, false, Bv, (short)0, acc, false, false);
    }
    float* Sq = Sbuf + q * 4096;
#pragma unroll
    for (int i = 0; i < 8; ++i) Sq[(i + 8 * half) * 256 + col] = acc[i];
  }
#else
  for (int e = tid; e < 5 * 4096; e += NTHREADS) {
    const int q = e / 4096, r = e % 4096;
    const int mm = r >> 8, col = r & 255;
    const float* p = hpass + q * 6144 + mm * 256 + col;
    float s = 0.f;
#pragma unroll
    for (int k = 0; k < 9; ++k) s += p[k * 256];
    Sbuf[e] = s;
  }
#endif
  __syncthreads();

  // ---- Stage D: per-voxel cc + block partial sum ----
  float lsum = 0.f;
  const float inv = 1.0f / 729.0f;
  for (int e = tid; e < 4096; e += NTHREADS) {
    const float S0 = Sbuf[e];
    const float S1 = Sbuf[4096 + e];
    const float S2 = Sbuf[2 * 4096 + e];
    const float S3 = Sbuf[3 * 4096 + e];
    const float S4 = Sbuf[4 * 4096 + e];
    const float mu1 = S0 * inv, mu2 = S1 * inv;
    const float s1  = S2 * inv - mu1 * mu1;
    const float s2  = S3 * inv - mu2 * mu2;
    const float s12 = S4 * inv - mu1 * mu2;
    lsum += (s12 * s12) / (s1 * s2 + 1e-5f);
  }
  float* red = smem;   // region1 free after stage C barrier
  red[tid] = lsum;
  __syncthreads();
#pragma unroll
  for (int s = NTHREADS / 2; s > 0; s >>= 1) {
    if (tid < s) red[tid] += red[tid + s];
    __syncthreads();
  }
  if (tid == 0) {
    const int bid = (blockIdx.z * gridDim.y + blockIdx.y) * gridDim.x + blockIdx.x;
    partial[bid] = red[0];
  }
}

__global__ __launch_bounds__(NTHREADS) void ncc_reduce_kernel(
    const float* __restrict__ partial, float* __restrict__ out, int nb) {
  __shared__ float red[NTHREADS];
  const int tid = threadIdx.x;
  float s = 0.f;
  for (int i = tid; i < nb; i += NTHREADS) s += partial[i];
  red[tid] = s;
  __syncthreads();
#pragma unroll
  for (int k = NTHREADS / 2; k > 0; k >>= 1) {
    if (tid < k) red[tid] += red[tid + k];
    __syncthreads();
  }
  if (tid == 0)
    out[0] = -red[0] / (float)((long)BB * DD * HH * WW);   // -mean(cc)
}

extern "C" void kernel_launch(void* const* d_in, const int* in_sizes, int n_in,
                              void* d_out, int out_size, void* d_ws, size_t ws_size,
                              hipStream_t stream) {
  (void)in_sizes; (void)n_in; (void)out_size; (void)ws_size;
  const float* I = (const float*)d_in[0];   // y_true
  const float* J = (const float*)d_in[1];   // y_pred
  float* partial = (float*)d_ws;
  float* out = (float*)d_out;

  dim3 grid(WW / TQ, HH / TQ, BB * (DD / TQ));   // 10 x 12 x 20 = 2400 blocks
  const int nblocks = grid.x * grid.y * grid.z;
  ncc_tile_kernel<<<grid, NTHREADS, SMEM_FLOATS * sizeof(float), stream>>>(I, J, partial);
  ncc_reduce_kernel<<<1, NTHREADS, 0, stream>>>(partial, out, nblocks);
}